// EGAT_21303037788659
// MI455X (gfx1250) — compile-verified
//
#include <hip/hip_runtime.h>
#include <hip/hip_bf16.h>
#include <math.h>

#define N_SRC 50000
#define N_DST 50000
#define NE    800000
#define IN_NODE 128
#define OUT_NODE 64
#define OUT_EDGE 16
#define H 4

typedef __attribute__((ext_vector_type(2))) float v2f;
typedef __attribute__((ext_vector_type(8))) float v8f;

// ---------------------------------------------------------------------------
// Generic fill
// ---------------------------------------------------------------------------
__global__ void egat_fill_f32(float* __restrict__ p, float v, int n) {
    int i = blockIdx.x * blockDim.x + threadIdx.x;
    if (i < n) p[i] = v;
}

// ---------------------------------------------------------------------------
// wsum[n] = sum_r W_fij[r, n]   (W_fij is 16 x 64; efeats@W_fij == reward*wsum)
// ---------------------------------------------------------------------------
__global__ void egat_wsum(const float* __restrict__ W_fij, float* __restrict__ wsum) {
    int n = threadIdx.x;  // 64 threads
    float s = 0.f;
#pragma unroll
    for (int r = 0; r < 16; ++r) s += W_fij[r * 64 + n];
    wsum[n] = s;
}

// ---------------------------------------------------------------------------
// Tiled GEMM Y = X @ W (+bias) using V_WMMA_F32_16X16X4_F32.
// One wave computes one 16x16 tile of Y. Block = 128 threads = 4 waves
// covering a 16(M) x 64(N) tile. K is a multiple of 4. Rows multiple of 16.
// A frag (16x4):  lanes 0-15 row M=lane, regs = {K, K+1}; lanes 16-31 = {K+2, K+3}
// B frag (4x16):  lanes 0-15 col N=lane, regs = {K, K+1}; lanes 16-31 = {K+2, K+3}
// C/D (16x16):    reg g: lanes 0-15 -> M=g, lanes 16-31 -> M=g+8; N = lane&15
// ---------------------------------------------------------------------------
__global__ void egat_proj_gemm(const float* __restrict__ X, const float* __restrict__ W,
                               const float* __restrict__ bias, float* __restrict__ Y,
                               int K, int Ncols) {
    const int wave  = threadIdx.x >> 5;
    const int lane  = threadIdx.x & 31;
    const int m_base = blockIdx.x * 16;
    const int n_base = blockIdx.y * 64 + wave * 16;
    if (n_base >= Ncols) return;            // wave-uniform, EXEC stays full

    const int l15  = lane & 15;
    const int koff = (lane < 16) ? 0 : 2;
    const float* __restrict__ arow = X + (size_t)(m_base + l15) * K + koff;
    const int ncol = n_base + l15;
    const float* __restrict__ bcol = W + (size_t)koff * Ncols + ncol;

    v8f acc = {};
    for (int k = 0; k < K; k += 4) {
        v2f a, b;
        a.x = arow[k];
        a.y = arow[k + 1];
        b.x = bcol[(size_t)k * Ncols];
        b.y = bcol[(size_t)(k + 1) * Ncols];
        acc = __builtin_amdgcn_wmma_f32_16x16x4_f32(
            /*neg_a=*/false, a, /*neg_b=*/false, b,
            /*c_mod=*/(short)0, acc, /*reuse_a=*/false, /*reuse_b=*/false);
    }

    const float bv = bias ? bias[ncol] : 0.0f;
    const int r0 = m_base + ((lane < 16) ? 0 : 8);
#pragma unroll
    for (int g = 0; g < 8; ++g)
        Y[(size_t)(r0 + g) * Ncols + ncol] = acc[g] + bv;
}

// ---------------------------------------------------------------------------
// Float atomic max via int/uint trick (works for mixed signs)
// ---------------------------------------------------------------------------
__device__ __forceinline__ void atomicMaxF(float* addr, float val) {
    if (val >= 0.0f)
        atomicMax((int*)addr, __float_as_int(val));
    else
        atomicMin((unsigned int*)addr, __float_as_uint(val));
}

// ---------------------------------------------------------------------------
// Per (edge, head): leaky-relu'd logit dot attn, store e_att, atomic-max into m
// ---------------------------------------------------------------------------
__global__ void egat_edge_logits(const float* __restrict__ f_ni, const float* __restrict__ f_nj,
                                 const float* __restrict__ reward,
                                 const int* __restrict__ src, const int* __restrict__ dst,
                                 const float* __restrict__ wsum, const float* __restrict__ b_e,
                                 const float* __restrict__ attn,
                                 float* __restrict__ e_att, float* __restrict__ m) {
    int t = blockIdx.x * blockDim.x + threadIdx.x;
    if (t >= NE * H) return;
    const int e = t >> 2, h = t & 3;
    const int sn = src[e], dn = dst[e];
    const float r = reward[e];
    const float* fi = f_ni + (size_t)sn * (H * OUT_EDGE) + h * OUT_EDGE;
    const float* fj = f_nj + (size_t)dn * (H * OUT_EDGE) + h * OUT_EDGE;
    const float* ws = wsum + h * OUT_EDGE;
    const float* be = b_e + h * OUT_EDGE;
    const float* at = attn + h * OUT_EDGE;
    float acc = 0.f;
#pragma unroll
    for (int f = 0; f < OUT_EDGE; ++f) {
        float v = fi[f] + fj[f] + r * ws[f] + be[f];
        v = (v > 0.f) ? v : 0.01f * v;
        acc += v * at[f];
    }
    e_att[t] = acc;
    atomicMaxF(&m[dn * H + h], acc);
}

// ---------------------------------------------------------------------------
// p = exp(e - m[dst]); accumulate s[dst] (softmax numerator + denominator)
// ---------------------------------------------------------------------------
__global__ void egat_softmax_num(const int* __restrict__ dst, const float* __restrict__ m,
                                 float* __restrict__ p /* in: e_att, out: exp */,
                                 float* __restrict__ s) {
    int t = blockIdx.x * blockDim.x + threadIdx.x;
    if (t >= NE * H) return;
    const int e = t >> 2, h = t & 3;
    const int dn = dst[e];
    float val = __expf(p[t] - m[dn * H + h]);
    p[t] = val;
    atomicAdd(&s[dn * H + h], val);
}

// ---------------------------------------------------------------------------
// One wave per edge. coef[h] = p/s / H; lane covers f and f+32.
// acc[dst, f] += sum_h h_src[src, h, f] * coef[h]
// ---------------------------------------------------------------------------
__global__ void egat_aggregate(const float* __restrict__ h_src, const float* __restrict__ p,
                               const float* __restrict__ s,
                               const int* __restrict__ src, const int* __restrict__ dst,
                               float* __restrict__ acc) {
    int wid = (blockIdx.x * blockDim.x + threadIdx.x) >> 5;
    int lane = threadIdx.x & 31;
    if (wid >= NE) return;
    const int sn = src[wid], dn = dst[wid];
    float coef[H];
#pragma unroll
    for (int h = 0; h < H; ++h) {
        float ss = s[dn * H + h];
        coef[h] = p[wid * H + h] / ((ss > 0.f) ? ss : 1.f) * 0.25f;
    }
    const float* hs = h_src + (size_t)sn * (H * OUT_NODE);
#pragma unroll
    for (int fo = 0; fo < 2; ++fo) {
        const int f = lane + fo * 32;
        float v = 0.f;
#pragma unroll
        for (int h = 0; h < H; ++h) v += hs[h * OUT_NODE + f] * coef[h];
        atomicAdd(&acc[(size_t)dn * OUT_NODE + f], v);
    }
}

// ---------------------------------------------------------------------------
// In-place ReLU on the accumulated output
// ---------------------------------------------------------------------------
__global__ void egat_relu(float* __restrict__ out, int n) {
    int i = blockIdx.x * blockDim.x + threadIdx.x;
    if (i < n) { float v = out[i]; out[i] = (v > 0.f) ? v : 0.f; }
}

// ---------------------------------------------------------------------------
extern "C" void kernel_launch(void* const* d_in, const int* in_sizes, int n_in,
                              void* d_out, int out_size, void* d_ws, size_t ws_size,
                              hipStream_t stream) {
    const float* nfeats    = (const float*)d_in[0];
    const float* dst_feats = (const float*)d_in[1];
    const float* reward    = (const float*)d_in[2];
    const int*   src       = (const int*)d_in[3];
    const int*   dst       = (const int*)d_in[4];
    const float* W_ns      = (const float*)d_in[5];
    const float* b_ns      = (const float*)d_in[6];
    const float* W_ni      = (const float*)d_in[7];
    const float* W_nj      = (const float*)d_in[8];
    const float* W_fij     = (const float*)d_in[9];
    const float* attn      = (const float*)d_in[10];
    const float* b_e       = (const float*)d_in[11];
    float* out = (float*)d_out;   // (N_DST, 64)

    // ---- workspace carve-up (all f32) ----
    char* ws = (char*)d_ws;
    size_t off = 0;
    auto carve = [&](size_t nfloats) { float* p = (float*)(ws + off);
                                       off += ((nfloats * 4 + 255) & ~(size_t)255); return p; };
    float* h_src = carve((size_t)N_SRC * H * OUT_NODE);   // 51.2 MB
    float* f_ni  = carve((size_t)N_SRC * H * OUT_EDGE);   // 12.8 MB
    float* f_nj  = carve((size_t)N_DST * H * OUT_EDGE);   // 12.8 MB
    float* e_att = carve((size_t)NE * H);                 // 12.8 MB (reused as p)
    float* m     = carve((size_t)N_DST * H);              // 0.8 MB
    float* ssum  = carve((size_t)N_DST * H);              // 0.8 MB
    float* wsum  = carve(64);

    // ---- init ----
    {
        int n = N_DST * H;
        egat_fill_f32<<<(n + 255) / 256, 256, 0, stream>>>(m, -3.0e38f, n);
        egat_fill_f32<<<(n + 255) / 256, 256, 0, stream>>>(ssum, 0.0f, n);
        int no = N_DST * OUT_NODE;
        egat_fill_f32<<<(no + 255) / 256, 256, 0, stream>>>(out, 0.0f, no);
    }
    egat_wsum<<<1, 64, 0, stream>>>(W_fij, wsum);

    // ---- node projections via WMMA (rows 50000 = 3125 * 16) ----
    {
        dim3 blk(128);
        dim3 g_ns(N_SRC / 16, (H * OUT_NODE) / 64);   // (3125, 4)
        egat_proj_gemm<<<g_ns, blk, 0, stream>>>(nfeats, W_ns, b_ns, h_src,
                                                 IN_NODE, H * OUT_NODE);
        dim3 g_ni(N_SRC / 16, (H * OUT_EDGE) / 64);   // (3125, 1)
        egat_proj_gemm<<<g_ni, blk, 0, stream>>>(nfeats, W_ni, nullptr, f_ni,
                                                 IN_NODE, H * OUT_EDGE);
        dim3 g_nj(N_DST / 16, (H * OUT_EDGE) / 64);
        egat_proj_gemm<<<g_nj, blk, 0, stream>>>(dst_feats, W_nj, nullptr, f_nj,
                                                 IN_NODE, H * OUT_EDGE);
    }

    // ---- edge attention ----
    {
        int n = NE * H;
        egat_edge_logits<<<(n + 255) / 256, 256, 0, stream>>>(
            f_ni, f_nj, reward, src, dst, wsum, b_e, attn, e_att, m);
        egat_softmax_num<<<(n + 255) / 256, 256, 0, stream>>>(dst, m, e_att, ssum);
    }

    // ---- weighted aggregation: one wave per edge, 8 edges per block ----
    egat_aggregate<<<(NE * 32 + 255) / 256, 256, 0, stream>>>(
        h_src, e_att, ssum, src, dst, out);

    // ---- final ReLU ----
    {
        int no = N_DST * OUT_NODE;
        egat_relu<<<(no + 255) / 256, 256, 0, stream>>>(out, no);
    }
}